// ObjCondensationLoss_42700564857097
// MI455X (gfx1250) — compile-verified
//
#include <hip/hip_runtime.h>
#include <hip/hip_bf16.h>
#include <math.h>

#define N_HIT   50000
#define N_TRUE  512
#define N_EDGE  40000
#define DD      8
#define N_GROUP (N_HIT / 16)                                  // 3125 hit groups of 16
#define WPB     8                                             // waves per block
#define MAIN_BLOCKS ((N_GROUP + WPB - 1) / WPB)               // 391

typedef float v2f __attribute__((ext_vector_type(2)));
typedef float v8f __attribute__((ext_vector_type(8)));
typedef unsigned v4u __attribute__((ext_vector_type(4)));
typedef int      v4i __attribute__((ext_vector_type(4)));
typedef int      v8i __attribute__((ext_vector_type(8)));

// ---- workspace layout in 32-bit words ----
#define WS_FC      0            // 512 floats   (f_centers, via uint atomicMax; f>0)
#define WS_CENTERS 512          // 512 ints     (center hit id per particle, init -1)
#define WS_XC      1024         // 4096 floats  (gathered center coords, row-major 512x8)
#define WS_CNQ     5120         // 1024 floats  (per particle: {||c||^2, q_center}) -- contiguous after XC
#define WS_PART    6144         // MAIN_BLOCKS floats (block partial sums of v)
#define WS_MASK    8192         // N_HIT*16 uints (512-bit membership row per hit)
// total = (8192 + 800000)*4 bytes ~= 3.23 MB

#define STAGE_WORDS (N_TRUE * DD + N_TRUE * 2)    // 5120 dwords staged to LDS (xc + cnq)

__global__ void k_init(unsigned* ws) {
  int i = blockIdx.x * blockDim.x + threadIdx.x;
  if (i < N_HIT * 16) ws[WS_MASK + i] = 0u;
  if (i < N_TRUE) {
    ws[WS_FC + i] = 0u;                 // bits of 0.0f
    ((int*)ws)[WS_CENTERS + i] = -1;
  }
}

// pass 1: per-particle max filter value + membership bitset
__global__ void k_edges1(const int* __restrict__ e_h, const int* __restrict__ e_p,
                         const float* __restrict__ f, unsigned* ws) {
  int e = blockIdx.x * blockDim.x + threadIdx.x;
  if (e >= N_EDGE) return;
  int h = e_h[e], p = e_p[e];
  atomicMax(&ws[WS_FC + p], __float_as_uint(f[h]));              // f > 0 -> uint order == float order
  atomicOr(&ws[WS_MASK + h * 16 + (p >> 5)], 1u << (p & 31));
}

// pass 2: among edges attaining the max, keep largest hit id (reference argmax rule)
__global__ void k_edges2(const int* __restrict__ e_h, const int* __restrict__ e_p,
                         const float* __restrict__ f, unsigned* ws) {
  int e = blockIdx.x * blockDim.x + threadIdx.x;
  if (e >= N_EDGE) return;
  int h = e_h[e], p = e_p[e];
  float fc = __uint_as_float(ws[WS_FC + p]);
  if (f[h] == fc) atomicMax(&((int*)ws)[WS_CENTERS + p], h);     // exact-bit equality, as reference
}

// gather center coords, center norms, q[center]
__global__ void k_centers(const float* __restrict__ x, const float* __restrict__ f,
                          unsigned* wsu) {
  int k = blockIdx.x * blockDim.x + threadIdx.x;
  if (k >= N_TRUE) return;
  float* ws = (float*)wsu;
  int c = ((int*)wsu)[WS_CENTERS + k];          // >= 0 guaranteed (every particle has a hit)
  float cn = 0.f;
#pragma unroll
  for (int d = 0; d < DD; ++d) {
    float v = x[c * DD + d];
    ws[WS_XC + k * DD + d] = v;
    cn = fmaf(v, v, cn);
  }
  float a = atanhf(f[c]);
  ws[WS_CNQ + 2 * k]     = cn;
  ws[WS_CNQ + 2 * k + 1] = fmaf(a, a, 0.5f);    // q = atanh(f)^2 + Q_MIN
}

// main pairwise kernel: one wave = 16 hits x all 512 particles, fp32 WMMA K=8 (2 x 16x16x4)
__global__ __launch_bounds__(256) void k_main(const float* __restrict__ x,
                                              const float* __restrict__ f,
                                              unsigned* __restrict__ wsu) {
  // single shared block so the TDM descriptor's lds_addr==0 targets the staging region
  __shared__ float smem[STAGE_WORDS + WPB];     // xc(4096) | cnq(1024) | s_red(8)
  float* s_xc  = smem;
  float* s_cnq = smem + N_TRUE * DD;
  float* s_red = smem + STAGE_WORDS;
  const float* wsf = (const float*)wsu;

  // ---- Tensor Data Mover: DMA the 20KB staging block (xc+cnq, contiguous in ws) to LDS.
  // 1-D D#: data_size=4B, tile_dim0=tensor_dim0=stride=5120 dwords, lds_addr=0.
  // Issued by wave 0 only (TDM ignores EXEC; SGPR descriptor), tracked by TENSORcnt.
  if ((threadIdx.x >> 5) == 0) {
    unsigned long long gaddr = (unsigned long long)(wsf + WS_XC);
    v4u g0;
    g0[0] = 1u;                                         // count=1, user mode, no gather
    g0[1] = 0u;                                         // lds_addr = 0
    g0[2] = (unsigned)(gaddr & 0xFFFFFFFFull);          // global_addr[31:0]
    g0[3] = (unsigned)((gaddr >> 32) & 0x1FFFFFFull) | 0x80000000u;  // addr[56:32] | type=2
    v8i g1;
    g1[0] = 0x20000;                                    // wg_mask=0, data_size=2 (4B)
    g1[1] = (int)(STAGE_WORDS << 16);                   // tensor_dim0 lo16 in [31:16]
    g1[2] = (int)((STAGE_WORDS >> 16) | (1u << 16));    // tensor_dim0 hi16, tensor_dim1=1
    g1[3] = (int)(STAGE_WORDS << 16);                   // tile_dim0 = 5120
    g1[4] = 0;                                          // tile_dim1/2 = 0 (1-D tile)
    g1[5] = STAGE_WORDS;                                // tensor_dim0_stride lo32
    g1[6] = 0;
    g1[7] = 0;
    v4i gz4 = {0, 0, 0, 0};
    v8i gz8 = {0, 0, 0, 0, 0, 0, 0, 0};
    __builtin_amdgcn_tensor_load_to_lds(g0, g1, gz4, gz4, gz8, 0);  // 6-arg form (clang-23)
    __builtin_amdgcn_s_wait_tensorcnt(0);
  }
  // Belt-and-braces: redundant cooperative load of the same bytes (overwrites TDM result,
  // so correctness never depends on the D# bit-packing).
  for (int i = threadIdx.x; i < STAGE_WORDS; i += 256) smem[i] = wsf[WS_XC + i];
  __syncthreads();

  const int lane  = threadIdx.x & 31;
  const int wave  = threadIdx.x >> 5;
  const int group = blockIdx.x * WPB + wave;        // wave-uniform -> EXEC all-1s inside branch
  float acc = 0.f;

  if (group < N_GROUP) {
    const int n16     = lane & 15;                  // hit column N within tile
    const int khi     = lane >> 4;                  // lane half selects K pair per WMMA layout
    const int hit     = group * 16 + n16;
    const int halfOff = khi << 3;                   // C rows 0..7 vs 8..15
    const int kk      = khi << 1;                   // A/B K columns {0,1}/{2,3} (+4 for chunk 1)

    const float4* xr = (const float4*)(x + hit * DD);
    float4 v0 = xr[0];
    float4 v1 = xr[1];
    float xn = v0.x*v0.x + v0.y*v0.y + v0.z*v0.z + v0.w*v0.w +
               v1.x*v1.x + v1.y*v1.y + v1.z*v1.z + v1.w*v1.w;
    // B matrix (8 x 16 hits): VGPR0 = row K=kk, VGPR1 = row K=kk+1, per ISA layout
    v2f b0 = khi ? v2f{v0.z, v0.w} : v2f{v0.x, v0.y};
    v2f b1 = khi ? v2f{v1.z, v1.w} : v2f{v1.x, v1.y};

    float at = atanhf(f[hit]);
    float q  = fmaf(at, at, 0.5f);

    // 512-bit membership row: 4 x b128 loads, word select resolved at compile time
    const uint4* mrow4 = (const uint4*)(wsu + WS_MASK + (size_t)hit * 16);

    for (int w2 = 0; w2 < 4; ++w2) {               // 4 x uint4 = 16 mask words = 32 tiles
      const uint4 cur = mrow4[w2];
#pragma unroll
      for (int wi = 0; wi < 4; ++wi) {
        const unsigned mw = (wi == 0) ? cur.x : (wi == 1) ? cur.y : (wi == 2) ? cur.z : cur.w;
#pragma unroll
        for (int hh = 0; hh < 2; ++hh) {
          const int pbase = ((w2 * 4 + wi) * 2 + hh) * 16;
          const v2f a0 = *(const v2f*)&s_xc[(pbase + n16) * DD + kk];      // A: 16 centers x K{0..3}
          const v2f a1 = *(const v2f*)&s_xc[(pbase + n16) * DD + kk + 4];  // A: K{4..7}
          v8f c = {0.f, 0.f, 0.f, 0.f, 0.f, 0.f, 0.f, 0.f};
          c = __builtin_amdgcn_wmma_f32_16x16x4_f32(false, a0, false, b0, (short)0, c, false, false);
          c = __builtin_amdgcn_wmma_f32_16x16x4_f32(false, a1, false, b1, (short)0, c, false, false);
          unsigned msh = mw >> (hh * 16 + halfOff);
#pragma unroll
          for (int r = 0; r < 8; ++r) {
            const int p     = pbase + halfOff + r;       // particle id for C row
            const float cn  = s_cnq[2 * p];
            const float qc  = s_cnq[2 * p + 1];
            const float dist = fmaf(-2.f, c[r], xn + cn);
            const float vv   = ((msh >> r) & 1u) ? dist : fmaxf(1.f - dist, 0.f);
            acc = fmaf(vv, qc, acc);
          }
        }
      }
    }
    acc *= q;   // outer q_i weight
  }

  // wave32 reduction, then block partial (deterministic: no float atomics)
#pragma unroll
  for (int off = 16; off > 0; off >>= 1) acc += __shfl_down(acc, off, 32);
  if (lane == 0) s_red[wave] = acc;
  __syncthreads();
  if (threadIdx.x == 0) {
    float s = 0.f;
#pragma unroll
    for (int i = 0; i < WPB; ++i) s += s_red[i];
    ((float*)wsu)[WS_PART + blockIdx.x] = s;
  }
}

// single-block deterministic finalize: background sums + tree reductions
__global__ __launch_bounds__(512) void k_finalize(const float* __restrict__ f,
                                                  const int* __restrict__ y,
                                                  const unsigned* __restrict__ wsu,
                                                  float* __restrict__ out) {
  __shared__ float rb[512], rp[512], rf[512];
  __shared__ int   rn[512];
  const float* ws = (const float*)wsu;
  int t = threadIdx.x;
  float bs = 0.f; int nb = 0;
  for (int i = t; i < N_HIT; i += 512) {
    if (y[i] == -1) { bs += f[i]; nb += 1; }
  }
  rb[t] = bs;
  rp[t] = (t < MAIN_BLOCKS) ? ws[WS_PART + t] : 0.f;
  rf[t] = ws[WS_FC + t];
  rn[t] = nb;
  __syncthreads();
  for (int s = 256; s > 0; s >>= 1) {
    if (t < s) { rb[t] += rb[t + s]; rp[t] += rp[t + s]; rf[t] += rf[t + s]; rn[t] += rn[t + s]; }
    __syncthreads();
  }
  if (t == 0) {
    float b1 = 1.f - rf[0] / (float)N_TRUE;
    float b2 = rb[0] / (float)rn[0];          // S_B = 1
    out[0] = b1 + b2;
    out[1] = rp[0] / (float)N_HIT;
  }
}

extern "C" void kernel_launch(void* const* d_in, const int* in_sizes, int n_in,
                              void* d_out, int out_size, void* d_ws, size_t ws_size,
                              hipStream_t stream) {
  (void)in_sizes; (void)n_in; (void)out_size; (void)ws_size;
  const float* x  = (const float*)d_in[0];
  const float* f  = (const float*)d_in[1];
  const int*   y  = (const int*)d_in[2];
  const int*   eh = (const int*)d_in[3];
  const int*   ep = (const int*)d_in[4];
  float*    out = (float*)d_out;
  unsigned* ws  = (unsigned*)d_ws;

  k_init    <<<(N_HIT * 16 + 255) / 256, 256, 0, stream>>>(ws);
  k_edges1  <<<(N_EDGE + 255) / 256,     256, 0, stream>>>(eh, ep, f, ws);
  k_edges2  <<<(N_EDGE + 255) / 256,     256, 0, stream>>>(eh, ep, f, ws);
  k_centers <<<(N_TRUE + 255) / 256,     256, 0, stream>>>(x, f, ws);
  k_main    <<<MAIN_BLOCKS,              256, 0, stream>>>(x, f, ws);
  k_finalize<<<1,                        512, 0, stream>>>(f, y, ws, out);
}